// PyGGATModel_48893907698331
// MI455X (gfx1250) — compile-verified
//
#include <hip/hip_runtime.h>
#include <hip/hip_bf16.h>
#include <math.h>

// ---------------------------------------------------------------------------
// GAT (2 layers, H=8 heads, F=32) for MI455X / gfx1250.
// GEMMs use V_WMMA_F32_16X16X4_F32 (exact fp32, wave32 WMMA).
// Edge phase uses hardware f32 atomics (add / sign-split max).
// ---------------------------------------------------------------------------

typedef __attribute__((ext_vector_type(2))) float v2f;
typedef __attribute__((ext_vector_type(8))) float v8f;

#define HEADS 8
#define FDIM  32
#define HF    256   // HEADS*FDIM
#define NEG_SLOPE 0.2f

// ---- float atomic max via monotonic int encoding (real HW atomics) --------
__device__ __forceinline__ void atomicMaxF32(float* addr, float value) {
  if (value >= 0.0f) {
    atomicMax((int*)addr, __float_as_int(value));
  } else {
    atomicMin((unsigned int*)addr, __float_as_uint(value));
  }
}

__device__ __forceinline__ float fAtomicAdd(float* addr, float value) {
  // lowers to global_atomic_add_f32 on gfx1250
  return unsafeAtomicAdd(addr, value);
}

// ---------------------------------------------------------------------------
// GEMM: Hout[M, 256] = X[M, K] @ W[K, 256]     (K multiple of 32)
// Block = 256 threads = 8 waves. Each block computes a 16-row x 256-col strip;
// wave w owns columns [32w, 32w+32) as two 16x16 WMMA tiles.
// K is streamed through LDS in chunks of 32 (34 KB static LDS).
//
// f32 WMMA operand layout (16x16x4):
//   A 16x4 : lanes 0-15 -> M=lane, VGPR{0,1}=K{0,1}; lanes 16-31 -> K{2,3}
//   B 4x16 : VGPR0 lanes0-15 K=0 / lanes16-31 K=2; VGPR1 K=1 / K=3; N=lane&15
//   C/D    : VGPR i -> M = i + (lane>=16 ? 8 : 0), N = lane&15
// ---------------------------------------------------------------------------
__global__ void gat_gemm_wmma(const float* __restrict__ X,
                              const float* __restrict__ W,
                              float* __restrict__ Hout,
                              int M, int K) {
  __shared__ float Wl[32 * HF];   // 32 KB
  __shared__ float Xl[16 * 32];   //  2 KB

  const int tid   = threadIdx.x;
  const int row0  = blockIdx.x * 16;
  const int wave  = tid >> 5;
  const int lane  = tid & 31;
  const int lrow  = lane & 15;
  const int khalf = (lane >> 4) << 1;   // 0 or 2
  const int col0  = wave * 32;

  v8f acc0 = {};
  v8f acc1 = {};

  for (int kc = 0; kc < K; kc += 32) {
    // cooperative loads
    for (int i = tid; i < 32 * HF; i += 256) {
      Wl[i] = W[(size_t)(kc + (i >> 8)) * HF + (i & 255)];
    }
    for (int i = tid; i < 16 * 32; i += 256) {
      int r = i >> 5, c = i & 31;
      Xl[i] = (row0 + r < M) ? X[(size_t)(row0 + r) * K + kc + c] : 0.0f;
    }
    __syncthreads();

    #pragma unroll
    for (int k0 = 0; k0 < 32; k0 += 4) {
      v2f a, b0, b1;
      a.x  = Xl[lrow * 32 + k0 + khalf + 0];
      a.y  = Xl[lrow * 32 + k0 + khalf + 1];
      b0.x = Wl[(k0 + khalf + 0) * HF + col0 + lrow];
      b0.y = Wl[(k0 + khalf + 1) * HF + col0 + lrow];
      b1.x = Wl[(k0 + khalf + 0) * HF + col0 + 16 + lrow];
      b1.y = Wl[(k0 + khalf + 1) * HF + col0 + 16 + lrow];
      acc0 = __builtin_amdgcn_wmma_f32_16x16x4_f32(false, a, false, b0,
                                                   (short)0, acc0, false, false);
      acc1 = __builtin_amdgcn_wmma_f32_16x16x4_f32(false, a, false, b1,
                                                   (short)0, acc1, false, false);
    }
    __syncthreads();
  }

  const int mo = (lane >> 4) << 3;   // 0 or 8
  #pragma unroll
  for (int i = 0; i < 8; ++i) {
    int r = row0 + mo + i;
    if (r < M) {
      size_t base = (size_t)r * HF + col0 + lrow;
      Hout[base]      = acc0[i];
      Hout[base + 16] = acc1[i];
    }
  }
}

// ---------------------------------------------------------------------------
// Per-(node, head) attention logits: s = <h, a_src>, d = <h, a_dst>
// ---------------------------------------------------------------------------
__global__ void gat_logits(const float* __restrict__ Hb,
                           const float* __restrict__ a_src,
                           const float* __restrict__ a_dst,
                           float* __restrict__ sv,
                           float* __restrict__ dv,
                           int N) {
  int i = blockIdx.x * blockDim.x + threadIdx.x;   // (n, h)
  if (i >= N * HEADS) return;
  int n = i >> 3, h = i & 7;
  const float* hp = Hb + (size_t)n * HF + h * FDIM;
  const float* as = a_src + h * FDIM;
  const float* ad = a_dst + h * FDIM;
  float s = 0.0f, d = 0.0f;
  #pragma unroll
  for (int f = 0; f < FDIM; ++f) {
    float v = hp[f];
    s += v * as[f];
    d += v * ad[f];
  }
  sv[i] = s;
  dv[i] = d;
}

// ---------------------------------------------------------------------------
// Init m = -inf, z = 0, agg = 0
// ---------------------------------------------------------------------------
__global__ void gat_init(float* __restrict__ m, float* __restrict__ z,
                         float* __restrict__ agg, int N) {
  int i = blockIdx.x * blockDim.x + threadIdx.x;
  if (i < N * HF) agg[i] = 0.0f;
  if (i < N * HEADS) { m[i] = -INFINITY; z[i] = 0.0f; }
}

// ---------------------------------------------------------------------------
// Edge pass A: m[dst,h] = max over edges of leaky_relu(s[src,h] + d[dst,h])
// Edge ids >= E are synthesized self-loops (src = dst = id - E).
// ---------------------------------------------------------------------------
__global__ void gat_edge_max(const int* __restrict__ ei,
                             const float* __restrict__ sv,
                             const float* __restrict__ dv,
                             float* __restrict__ m,
                             int E, int N) {
  int i = blockIdx.x * blockDim.x + threadIdx.x;   // (edge, h)
  if (i >= (E + N) * HEADS) return;
  int e = i >> 3, h = i & 7;
  int src, dst;
  if (e < E) { src = ei[e]; dst = ei[E + e]; } else { src = dst = e - E; }
  float x = sv[src * HEADS + h] + dv[dst * HEADS + h];
  x = (x > 0.0f) ? x : NEG_SLOPE * x;
  atomicMaxF32(m + dst * HEADS + h, x);
}

// ---------------------------------------------------------------------------
// Edge pass B: ex = exp(e - m[dst]); store ex; z[dst,h] += ex
// ---------------------------------------------------------------------------
__global__ void gat_edge_exp(const int* __restrict__ ei,
                             const float* __restrict__ sv,
                             const float* __restrict__ dv,
                             const float* __restrict__ m,
                             float* __restrict__ exb,
                             float* __restrict__ z,
                             int E, int N) {
  int i = blockIdx.x * blockDim.x + threadIdx.x;   // (edge, h)
  if (i >= (E + N) * HEADS) return;
  int e = i >> 3, h = i & 7;
  int src, dst;
  if (e < E) { src = ei[e]; dst = ei[E + e]; } else { src = dst = e - E; }
  float x = sv[src * HEADS + h] + dv[dst * HEADS + h];
  x = (x > 0.0f) ? x : NEG_SLOPE * x;
  float ex = __expf(x - m[dst * HEADS + h]);
  exb[i] = ex;
  fAtomicAdd(z + dst * HEADS + h, ex);
}

// ---------------------------------------------------------------------------
// Edge pass C (bandwidth dominant): agg[dst, j] += ex[edge, j/32] * h[src, j]
// One thread per (edge, j in 0..255): 1 KB contiguous per edge, coalesced.
// ---------------------------------------------------------------------------
__global__ void gat_edge_agg(const int* __restrict__ ei,
                             const float* __restrict__ exb,
                             const float* __restrict__ Hb,
                             float* __restrict__ agg,
                             int E, int N) {
  int i = blockIdx.x * blockDim.x + threadIdx.x;   // (edge, j)
  int e = i >> 8, j = i & 255;
  if (e >= E + N) return;
  int src, dst;
  if (e < E) { src = ei[e]; dst = ei[E + e]; } else { src = dst = e - E; }
  float ex = exb[e * HEADS + (j >> 5)];
  fAtomicAdd(agg + (size_t)dst * HF + j, ex * Hb[(size_t)src * HF + j]);
}

// ---------------------------------------------------------------------------
// Finalize: out[n, f] = mean_h( agg[n,h,f] / z[n,h] ) + bias[f]  (opt. ReLU)
// ---------------------------------------------------------------------------
__global__ void gat_finalize(const float* __restrict__ agg,
                             const float* __restrict__ z,
                             const float* __restrict__ bias,
                             float* __restrict__ out,
                             int N, int do_relu) {
  int i = blockIdx.x * blockDim.x + threadIdx.x;   // (n, f)
  if (i >= N * FDIM) return;
  int n = i >> 5, f = i & 31;
  float acc = 0.0f;
  #pragma unroll
  for (int h = 0; h < HEADS; ++h) {
    acc += agg[(size_t)n * HF + h * FDIM + f] / z[n * HEADS + h];
  }
  float v = acc * (1.0f / HEADS) + bias[f];
  if (do_relu) v = fmaxf(v, 0.0f);
  out[i] = v;
}

// ---------------------------------------------------------------------------
// Launch
// ---------------------------------------------------------------------------
static inline int ceil_div(int a, int b) { return (a + b - 1) / b; }

extern "C" void kernel_launch(void* const* d_in, const int* in_sizes, int n_in,
                              void* d_out, int out_size, void* d_ws, size_t ws_size,
                              hipStream_t stream) {
  const float* x   = (const float*)d_in[0];
  const int*   ei  = (const int*)  d_in[1];
  const float* W1  = (const float*)d_in[2];
  const float* as1 = (const float*)d_in[3];
  const float* ad1 = (const float*)d_in[4];
  const float* b1  = (const float*)d_in[5];
  const float* W2  = (const float*)d_in[6];
  const float* as2 = (const float*)d_in[7];
  const float* ad2 = (const float*)d_in[8];
  const float* b2  = (const float*)d_in[9];
  float* out = (float*)d_out;

  const int Din = 128;
  const int N   = in_sizes[0] / Din;
  const int E   = in_sizes[1] / 2;
  const int ET  = E + N;

  // workspace layout (floats)
  float* ws   = (float*)d_ws;
  size_t o    = 0;
  float* hbuf = ws + o; o += (size_t)N * HF;      // node features per head
  float* agg  = ws + o; o += (size_t)N * HF;      // unnormalized aggregation
  float* sv   = ws + o; o += (size_t)N * HEADS;   // source logits
  float* dv   = ws + o; o += (size_t)N * HEADS;   // dest logits
  float* mv   = ws + o; o += (size_t)N * HEADS;   // segment max
  float* zv   = ws + o; o += (size_t)N * HEADS;   // segment sum of exp
  float* exb  = ws + o; o += (size_t)ET * HEADS;  // per-edge exp values
  float* xmid = ws + o; o += (size_t)N * FDIM;    // layer-1 output
  (void)ws_size;

  const int BT = 256;
  const int gemm_blocks = ceil_div(N, 16);
  const int nh_blocks   = ceil_div(N * HEADS, BT);
  const int init_blocks = ceil_div(N * HF, BT);
  const int eh_blocks   = ceil_div(ET * HEADS, BT);
  const int ej_blocks   = ET;                     // ET * 256 threads / 256
  const int nf_blocks   = ceil_div(N * FDIM, BT);

  // ---------------- Layer 1 ----------------
  gat_gemm_wmma<<<gemm_blocks, BT, 0, stream>>>(x, W1, hbuf, N, Din);
  gat_logits  <<<nh_blocks,   BT, 0, stream>>>(hbuf, as1, ad1, sv, dv, N);
  gat_init    <<<init_blocks, BT, 0, stream>>>(mv, zv, agg, N);
  gat_edge_max<<<eh_blocks,   BT, 0, stream>>>(ei, sv, dv, mv, E, N);
  gat_edge_exp<<<eh_blocks,   BT, 0, stream>>>(ei, sv, dv, mv, exb, zv, E, N);
  gat_edge_agg<<<ej_blocks,   BT, 0, stream>>>(ei, exb, hbuf, agg, E, N);
  gat_finalize<<<nf_blocks,   BT, 0, stream>>>(agg, zv, b1, xmid, N, /*relu=*/1);

  // ---------------- Layer 2 ----------------
  gat_gemm_wmma<<<gemm_blocks, BT, 0, stream>>>(xmid, W2, hbuf, N, FDIM);
  gat_logits  <<<nh_blocks,   BT, 0, stream>>>(hbuf, as2, ad2, sv, dv, N);
  gat_init    <<<init_blocks, BT, 0, stream>>>(mv, zv, agg, N);
  gat_edge_max<<<eh_blocks,   BT, 0, stream>>>(ei, sv, dv, mv, E, N);
  gat_edge_exp<<<eh_blocks,   BT, 0, stream>>>(ei, sv, dv, mv, exb, zv, E, N);
  gat_edge_agg<<<ej_blocks,   BT, 0, stream>>>(ei, exb, hbuf, agg, E, N);
  gat_finalize<<<nf_blocks,   BT, 0, stream>>>(agg, zv, b2, out, N, /*relu=*/0);
}